// CC_44349832298987
// MI455X (gfx1250) — compile-verified
//
#include <hip/hip_runtime.h>

typedef __attribute__((ext_vector_type(4))) float f4;
typedef __attribute__((ext_vector_type(2))) float v2f;
typedef __attribute__((ext_vector_type(8))) float v8f;

#define THRESH 0.001f
#define NB 32
#define NL 128
#define NN 8192

// ---------------------------------------------------------------------------
// Stage 1: one workgroup (256 thr = 8 waves) per (b,l) row.
// Single pass over N=8192 fp32 elements of pre & label, 6 masked moments,
// wave32 shuffle reduction + LDS cross-wave reduction, finalize cc -> ws[row].
// Memory-bound: non-temporal b128 loads (read-once streaming, keep L2 clean).
// ---------------------------------------------------------------------------
__global__ __launch_bounds__(256) void cc_row_kernel(const float* __restrict__ pre,
                                                     const float* __restrict__ label,
                                                     float* __restrict__ cc /* [B*L] */) {
  const int row = blockIdx.x;                 // 0..4095  (b*128 + l)
  const long base = (long)row * NN;
  const int t = threadIdx.x;                  // 0..255

  const f4* p4 = (const f4*)(pre + base);
  const f4* l4 = (const f4*)(label + base);

  float cnt = 0.f, sp = 0.f, sl = 0.f, spp = 0.f, sll = 0.f, spl = 0.f;

#pragma unroll
  for (int i = 0; i < NN / (256 * 4); ++i) {  // 8 iterations
    const int idx = i * 256 + t;              // coalesced f4 index
    f4 p = __builtin_nontemporal_load(&p4[idx]);
    f4 l = __builtin_nontemporal_load(&l4[idx]);
#pragma unroll
    for (int j = 0; j < 4; ++j) {
      const float pv = p[j];
      const float lv = l[j];
      const float m = ((__builtin_fabsf(pv) > THRESH) ||
                       (__builtin_fabsf(lv) > THRESH)) ? 1.0f : 0.0f;
      const float pm = pv * m;
      const float lm = lv * m;
      cnt += m;
      sp  += pm;
      sl  += lm;
      spp += pm * pv;   // p^2 * m   (m^2 == m)
      sll += lm * lv;   // l^2 * m
      spl += pm * lv;   // p*l * m
    }
  }

  // wave32 tree reduction
#pragma unroll
  for (int off = 16; off > 0; off >>= 1) {
    cnt += __shfl_down(cnt, off, 32);
    sp  += __shfl_down(sp,  off, 32);
    sl  += __shfl_down(sl,  off, 32);
    spp += __shfl_down(spp, off, 32);
    sll += __shfl_down(sll, off, 32);
    spl += __shfl_down(spl, off, 32);
  }

  __shared__ float red[8][6];
  const int lane = t & 31;
  const int wave = t >> 5;
  if (lane == 0) {
    red[wave][0] = cnt; red[wave][1] = sp;  red[wave][2] = sl;
    red[wave][3] = spp; red[wave][4] = sll; red[wave][5] = spl;
  }
  __syncthreads();

  if (t == 0) {
    float a0 = 0.f, a1 = 0.f, a2 = 0.f, a3 = 0.f, a4 = 0.f, a5 = 0.f;
#pragma unroll
    for (int w = 0; w < 8; ++w) {
      a0 += red[w][0]; a1 += red[w][1]; a2 += red[w][2];
      a3 += red[w][3]; a4 += red[w][4]; a5 += red[w][5];
    }
    const float inv = 1.0f / a0;
    const float cov = a5 - a1 * a2 * inv;
    const float vp  = a3 - a1 * a1 * inv;
    const float vl  = a4 - a2 * a2 * inv;
    cc[row] = cov / __builtin_sqrtf(vp * vl);
  }
}

// ---------------------------------------------------------------------------
// Stage 2: out[l] = sum_b cc[b,l]  ==  ones(1x32) @ CC(32x128).
// Done with V_WMMA_F32_16X16X4_F32: A = ones(16x4), B = 4x16 tile of CC,
// accumulate 8 K-chunks (32 = 8*4). Wave w owns columns [16w, 16w+16).
// B layout (32-bit B, 4x16): VGPR0 = K=0 (lanes 0-15) / K=2 (lanes 16-31),
// VGPR1 = K=1 / K=3, N = lane%16. D row M=0 lands in accVGPR0, lanes 0-15.
// Deterministic (no float atomics), engages the matrix pipe for free.
// ---------------------------------------------------------------------------
__global__ __launch_bounds__(256) void cc_bsum_wmma(const float* __restrict__ cc,
                                                    float* __restrict__ out) {
  const int t = threadIdx.x;
  const int lane = t & 31;
  const int wave = t >> 5;                 // 0..7 -> 16 columns each
  const int col = wave * 16 + (lane & 15); // l in [0,128)
  const int khalf = (lane >> 4) << 1;      // 0 for lanes 0-15, 2 for 16-31

  v8f acc = {};
  v2f a;  a[0] = 1.0f;  a[1] = 1.0f;       // ones A-matrix (layout irrelevant)

#pragma unroll
  for (int it = 0; it < 8; ++it) {
    const int k0 = it * 4 + khalf;         // b index base
    v2f b;
    b[0] = cc[(k0 + 0) * NL + col];        // B VGPR0: K = k0
    b[1] = cc[(k0 + 1) * NL + col];        // B VGPR1: K = k0 + 1
    acc = __builtin_amdgcn_wmma_f32_16x16x4_f32(
        /*neg_a=*/false, a, /*neg_b=*/false, b,
        /*c_mod=*/(short)0, acc, /*reuse_a=*/false, /*reuse_b=*/false);
  }

  if (lane < 16) out[col] = acc[0];        // M=0 row of D
}

extern "C" void kernel_launch(void* const* d_in, const int* in_sizes, int n_in,
                              void* d_out, int out_size, void* d_ws, size_t ws_size,
                              hipStream_t stream) {
  const float* pre   = (const float*)d_in[0];
  const float* label = (const float*)d_in[1];
  float* out = (float*)d_out;              // [128]
  float* cc  = (float*)d_ws;               // [32*128] scratch

  cc_row_kernel<<<NB * NL, 256, 0, stream>>>(pre, label, cc);
  cc_bsum_wmma<<<1, 256, 0, stream>>>(cc, out);
}